// MultiHeadGraphAttention_42494406426626
// MI455X (gfx1250) — compile-verified
//
#include <hip/hip_runtime.h>
#include <math.h>

typedef __attribute__((ext_vector_type(2))) float v2f;
typedef __attribute__((ext_vector_type(4))) float v4f;
typedef __attribute__((ext_vector_type(8))) float v8f;

#define HID 128
#define HEADS 8
#define HEAD_DIM 16

// ---- WMMA f32 16x16x4 fragment helpers (CDNA5 VGPR layouts, ISA 7.12.2) ----
__device__ __forceinline__ v2f a_frag(const float* A, int lda, int k, int lane) {
  int m = lane & 15;
  int kk = k + ((lane >> 4) << 1);
  v2f a;
  a.x = A[m * lda + kk];
  a.y = A[m * lda + kk + 1];
  return a;
}
__device__ __forceinline__ v2f b_frag(const float* B, int ldb, int k, int lane) {
  int n = lane & 15;
  int kk = k + ((lane >> 4) << 1);
  v2f b;
  b.x = B[kk * ldb + n];
  b.y = B[(kk + 1) * ldb + n];
  return b;
}
__device__ __forceinline__ v8f wmma_f32(v2f a, v2f b, v8f c) {
  return __builtin_amdgcn_wmma_f32_16x16x4_f32(false, a, false, b, (short)0, c,
                                               false, false);
}

// ---- CDNA5 async global->LDS copy (ASYNCcnt path, bypasses VGPRs) ----------
// Generic pointers to __shared__ are {shared_aperture_hi32, lds_offset32};
// the async instruction's VDST takes the 32-bit LDS byte offset.
__device__ __forceinline__ unsigned lds_off_u32(const void* p) {
  return (unsigned)(unsigned long long)p;
}
__device__ __forceinline__ void async_copy_b128(unsigned lds_off,
                                                const void* gaddr) {
  asm volatile("global_load_async_to_lds_b128 %0, %1, off" ::"v"(lds_off),
               "v"(gaddr)
               : "memory");
}
__device__ __forceinline__ void wait_asynccnt0() {
  asm volatile("s_wait_asynccnt 0" ::: "memory");
}

// sign-aware float atomic max (init with -inf); no CAS loop needed
__device__ __forceinline__ void atomicMaxF(float* addr, float val) {
  if (val >= 0.0f)
    atomicMax((int*)addr, __float_as_int(val));
  else
    atomicMin((unsigned int*)addr, __float_as_uint(val));
}

// ---------------------------------------------------------------------------
// Init: segment max -inf, denom 0, dst histogram 0, csr cursor 0
__global__ void init_kernel(float* __restrict__ mbuf, float* __restrict__ den,
                            int* __restrict__ hist, int* __restrict__ cursor,
                            long n8, long n) {
  long i = (long)blockIdx.x * blockDim.x + threadIdx.x;
  if (i < n8) {
    mbuf[i] = -__builtin_inff();
    den[i] = 0.0f;
  }
  if (i < n) {
    hist[i] = 0;
    cursor[i] = 0;
  }
}

// dst-degree histogram (1.6M cheap int atomics, replaces 205M f32 atomics)
__global__ void hist_kernel(const int* __restrict__ dst, int* __restrict__ hist,
                            long E) {
  long e = (long)blockIdx.x * blockDim.x + threadIdx.x;
  if (e >= E) return;
  atomicAdd(&hist[dst[e]], 1);
}

// single-workgroup chunked exclusive scan (N=100k -> ~98 chunks, trivial time)
__global__ __launch_bounds__(1024) void scan_kernel(const int* __restrict__ hist,
                                                    int* __restrict__ offs,
                                                    int n) {
  __shared__ int lds[1024];
  __shared__ int carry;
  int tid = threadIdx.x;
  if (tid == 0) carry = 0;
  __syncthreads();
  for (int base = 0; base < n; base += 1024) {
    int x = (base + tid < n) ? hist[base + tid] : 0;
    lds[tid] = x;
    __syncthreads();
    for (int off = 1; off < 1024; off <<= 1) {
      int t = (tid >= off) ? lds[tid - off] : 0;
      __syncthreads();
      lds[tid] += t;
      __syncthreads();
    }
    int incl = lds[tid];
    int c = carry;
    if (base + tid < n) offs[base + tid] = c + incl - x;  // exclusive
    __syncthreads();
    if (tid == 1023) carry = c + incl;
    __syncthreads();
  }
  if (tid == 0) offs[n] = carry;
}

// scatter edge ids into dst-CSR slots
__global__ void build_csr_kernel(const int* __restrict__ dst,
                                 int* __restrict__ cursor,
                                 const int* __restrict__ offs,
                                 int* __restrict__ eid, long E) {
  long e = (long)blockIdx.x * blockDim.x + threadIdx.x;
  if (e >= E) return;
  int d = dst[e];
  int p = atomicAdd(&cursor[d], 1);
  eid[offs[d] + p] = (int)e;
}

// ---------------------------------------------------------------------------
// Fused node pipeline: h = relu(nf@Wn+bn); Q/K/V = h@W{q,k,v}+b  (WMMA f32)
__global__ __launch_bounds__(128) void node_qkv_kernel(
    const float* __restrict__ nf, const float* __restrict__ Wn,
    const float* __restrict__ bn, const float* __restrict__ Wq,
    const float* __restrict__ bq, const float* __restrict__ Wk,
    const float* __restrict__ bk, const float* __restrict__ Wv,
    const float* __restrict__ bv, float* __restrict__ h, float* __restrict__ Q,
    float* __restrict__ K, float* __restrict__ V, int ntiles) {
  __shared__ __attribute__((aligned(16))) float nf_lds[4][16 * 64];
  __shared__ __attribute__((aligned(16))) float h_lds[4][16 * HID];
  int wave = threadIdx.x >> 5, lane = threadIdx.x & 31;
  int tile = blockIdx.x * 4 + wave;
  if (tile >= ntiles) return;
  size_t row0 = (size_t)tile * 16;

  // async-stage node_feat tile into LDS (8 x b128 per lane)
  for (int i = lane; i < 16 * 64 / 4; i += 32)
    async_copy_b128(lds_off_u32(&nf_lds[wave][i * 4]),
                    nf + row0 * 64 + (size_t)i * 4);
  wait_asynccnt0();

  int half = lane >> 4, n16 = lane & 15;

  // h = relu(nf @ Wn + bn) -> LDS + global
  for (int j = 0; j < 8; ++j) {
    v8f acc = {};
#pragma unroll
    for (int k = 0; k < 64; k += 4)
      acc = wmma_f32(a_frag(nf_lds[wave], 64, k, lane),
                     b_frag(Wn + j * 16, HID, k, lane), acc);
    int n = j * 16 + n16;
    float bb = bn[n];
#pragma unroll
    for (int r = 0; r < 8; ++r) {
      float v = fmaxf(acc[r] + bb, 0.0f);
      h_lds[wave][(r + half * 8) * HID + n] = v;
      h[(row0 + r + half * 8) * HID + n] = v;
    }
  }

  const float* Ws[3] = {Wq, Wk, Wv};
  const float* bs[3] = {bq, bk, bv};
  float* Os[3] = {Q, K, V};
  for (int p = 0; p < 3; ++p) {
    for (int j = 0; j < 8; ++j) {
      v8f acc = {};
#pragma unroll
      for (int k = 0; k < HID; k += 4)
        acc = wmma_f32(a_frag(h_lds[wave], HID, k, lane),
                       b_frag(Ws[p] + j * 16, HID, k, lane), acc);
      int n = j * 16 + n16;
      float bb = bs[p][n];
#pragma unroll
      for (int r = 0; r < 8; ++r)
        Os[p][(row0 + r + half * 8) * HID + n] = acc[r] + bb;
    }
  }
}

// ---------------------------------------------------------------------------
// Fused edge pipeline: e = relu(ef@We+be) (LDS only, never hits HBM),
// scores = QK/4 + e@Wea + bea (WMMA, 8->16 zero-padded cols),
// store [E,8] scores + atomic segment-max by src. 1 wave = 16 edges.
__global__ __launch_bounds__(128) void edge_score_kernel(
    const float* __restrict__ ef, const int* __restrict__ src,
    const int* __restrict__ dst, const float* __restrict__ We,
    const float* __restrict__ be, const float* __restrict__ Wea,
    const float* __restrict__ bea, const float* __restrict__ Q,
    const float* __restrict__ K, float* __restrict__ scores,
    float* __restrict__ mbuf, int ntiles) {
  __shared__ __attribute__((aligned(16))) float ef_lds[4][16 * 32];
  __shared__ __attribute__((aligned(16))) float e_lds[4][16 * HID];
  int wave = threadIdx.x >> 5, lane = threadIdx.x & 31;
  int tile = blockIdx.x * 4 + wave;
  if (tile >= ntiles) return;
  size_t e0 = (size_t)tile * 16;

  // async-stage edge_feat tile into LDS (4 x b128 per lane)
  for (int i = lane; i < 16 * 32 / 4; i += 32)
    async_copy_b128(lds_off_u32(&ef_lds[wave][i * 4]),
                    ef + e0 * 32 + (size_t)i * 4);
  wait_asynccnt0();

  int half = lane >> 4, n16 = lane & 15;

  for (int j = 0; j < 8; ++j) {
    v8f acc = {};
#pragma unroll
    for (int k = 0; k < 32; k += 4)
      acc = wmma_f32(a_frag(ef_lds[wave], 32, k, lane),
                     b_frag(We + j * 16, HID, k, lane), acc);
    int n = j * 16 + n16;
    float bb = be[n];
#pragma unroll
    for (int r = 0; r < 8; ++r)
      e_lds[wave][(r + half * 8) * HID + n] = fmaxf(acc[r] + bb, 0.0f);
  }

  // e @ Wea : B is 128x8, zero-pad to 16 columns
  v8f acc = {};
  for (int k = 0; k < HID; k += 4) {
    v2f a = a_frag(e_lds[wave], HID, k, lane);
    int kk = k + half * 2;
    v2f b;
    b.x = (n16 < HEADS) ? Wea[kk * HEADS + n16] : 0.0f;
    b.y = (n16 < HEADS) ? Wea[(kk + 1) * HEADS + n16] : 0.0f;
    acc = wmma_f32(a, b, acc);
  }

  // per-edge QK/sqrt(d) + bea; Q/K gathers are L2-resident (154MB < 192MB L2)
  if (n16 < HEADS) {
    int hd = n16;
    float bb = bea[hd];
#pragma unroll
    for (int r = 0; r < 8; ++r) {
      int m = r + half * 8;
      size_t eidx = e0 + m;
      int s = src[eidx], d = dst[eidx];
      const v4f* qp = (const v4f*)(Q + (size_t)s * HID + hd * HEAD_DIM);
      const v4f* kp = (const v4f*)(K + (size_t)d * HID + hd * HEAD_DIM);
      float dot = 0.0f;
#pragma unroll
      for (int t = 0; t < 4; ++t) {
        v4f q = qp[t], kk4 = kp[t];
        dot += q.x * kk4.x + q.y * kk4.y + q.z * kk4.z + q.w * kk4.w;
      }
      float sc = dot * 0.25f + acc[r] + bb;  // 1/sqrt(16) = 0.25
      scores[eidx * HEADS + hd] = sc;
      atomicMaxF(mbuf + (size_t)s * HEADS + hd, sc);
    }
  }
}

// ---------------------------------------------------------------------------
// ex = exp(score - m[src]); in-place; atomic segment-sum of denom by src
__global__ void softmax_exp_kernel(const int* __restrict__ src,
                                   float* __restrict__ scores,
                                   const float* __restrict__ mbuf,
                                   float* __restrict__ den, long EH) {
  long i = (long)blockIdx.x * blockDim.x + threadIdx.x;
  if (i >= EH) return;
  long e = i >> 3;
  int hd = (int)(i & 7);
  int s = src[e];
  float ex = __expf(scores[i] - mbuf[(size_t)s * HEADS + hd]);
  scores[i] = ex;
  atomicAdd(den + (size_t)s * HEADS + hd, ex);
}

// attn = ex / denom[src]  (streaming 51MB pass; den gathers L2-resident)
__global__ void normalize_kernel(const int* __restrict__ src,
                                 float* __restrict__ scores,
                                 const float* __restrict__ den, long EH) {
  long i = (long)blockIdx.x * blockDim.x + threadIdx.x;
  if (i >= EH) return;
  long e = i >> 3;
  int hd = (int)(i & 7);
  scores[i] = scores[i] / den[(size_t)src[e] * HEADS + hd];
}

// ---------------------------------------------------------------------------
// Atomic-free output accumulation: one wave per dst node walks its CSR edge
// list; lane covers 4 contiguous dims of one head (head = lane/4).
__global__ __launch_bounds__(256) void gather_out_kernel(
    const int* __restrict__ eid, const int* __restrict__ offs,
    const int* __restrict__ src, const float* __restrict__ V,
    const float* __restrict__ attn, float* __restrict__ accum, int Nn) {
  int wave = threadIdx.x >> 5, lane = threadIdx.x & 31;
  int node = blockIdx.x * 8 + wave;
  if (node >= Nn) return;
  int beg = offs[node], end = offs[node + 1];
  int hd = lane >> 2;
  int dsub = (lane & 3) * 4;
  v4f acc = {};
  for (int p = beg; p < end; ++p) {
    int e = eid[p];
    int s = src[e];
    float w = attn[(size_t)e * HEADS + hd];
    v4f vv = *(const v4f*)(V + (size_t)s * HID + hd * HEAD_DIM + dsub);
    acc.x += vv.x * w;
    acc.y += vv.y * w;
    acc.z += vv.z * w;
    acc.w += vv.w * w;
  }
  *(v4f*)(accum + (size_t)node * HID + lane * 4) = acc;
}

// ---------------------------------------------------------------------------
// out = accum @ Wo + bo + h  (accum lives in d_out; async-staged through LDS)
__global__ __launch_bounds__(128) void out_proj_kernel(
    const float* __restrict__ h, const float* __restrict__ Wo,
    const float* __restrict__ bo, float* __restrict__ out, int ntiles) {
  __shared__ __attribute__((aligned(16))) float a_lds[4][16 * HID];
  int wave = threadIdx.x >> 5, lane = threadIdx.x & 31;
  int tile = blockIdx.x * 4 + wave;
  if (tile >= ntiles) return;
  size_t row0 = (size_t)tile * 16;

  for (int i = lane; i < 16 * HID / 4; i += 32)
    async_copy_b128(lds_off_u32(&a_lds[wave][i * 4]),
                    out + row0 * HID + (size_t)i * 4);
  wait_asynccnt0();

  int half = lane >> 4, n16 = lane & 15;
  for (int j = 0; j < 8; ++j) {
    v8f acc = {};
#pragma unroll
    for (int k = 0; k < HID; k += 4)
      acc = wmma_f32(a_frag(a_lds[wave], HID, k, lane),
                     b_frag(Wo + j * 16, HID, k, lane), acc);
    int n = j * 16 + n16;
    float bb = bo[n];
#pragma unroll
    for (int r = 0; r < 8; ++r) {
      size_t row = row0 + r + half * 8;
      out[row * HID + n] = acc[r] + bb + h[row * HID + n];
    }
  }
}

// ---------------------------------------------------------------------------
extern "C" void kernel_launch(void* const* d_in, const int* in_sizes, int n_in,
                              void* d_out, int out_size, void* d_ws,
                              size_t ws_size, hipStream_t stream) {
  const float* nf = (const float*)d_in[0];
  const int* ei = (const int*)d_in[1];
  const float* ef = (const float*)d_in[2];
  const float* Wn = (const float*)d_in[3];
  const float* bn = (const float*)d_in[4];
  const float* We = (const float*)d_in[5];
  const float* be = (const float*)d_in[6];
  const float* Wq = (const float*)d_in[7];
  const float* bq = (const float*)d_in[8];
  const float* Wk = (const float*)d_in[9];
  const float* bk = (const float*)d_in[10];
  const float* Wv = (const float*)d_in[11];
  const float* bv = (const float*)d_in[12];
  const float* Wea = (const float*)d_in[13];
  const float* bea = (const float*)d_in[14];
  const float* Wo = (const float*)d_in[15];
  const float* bo = (const float*)d_in[16];

  int N = in_sizes[0] / 64;  // NODE_IN = 64
  int E = in_sizes[2] / 32;  // EDGE_IN = 32
  const int* srcI = ei;      // edge_index[0]
  const int* dstI = ei + E;  // edge_index[1]

  float* out = (float*)d_out;  // doubles as the pre-Wo accumulator
  float* ws = (float*)d_ws;
  size_t nh = (size_t)N * HID;
  float* h = ws;
  float* Qb = h + nh;
  float* Kb = Qb + nh;
  float* Vb = Kb + nh;
  float* mbuf = Vb + nh;              // N*8
  float* den = mbuf + (size_t)N * 8;  // N*8
  float* sc = den + (size_t)N * 8;    // E*8
  int* hist = (int*)(sc + (size_t)E * 8);
  int* cursor = hist + N;
  int* offs = cursor + N;  // N+1
  int* eid = offs + (N + 1);

  long n8 = (long)N * 8;
  long EH = (long)E * HEADS;

  {
    long g = (n8 + 255) / 256;
    init_kernel<<<dim3((unsigned)g), dim3(256), 0, stream>>>(mbuf, den, hist,
                                                             cursor, n8, N);
  }
  {
    long g = ((long)E + 255) / 256;
    hist_kernel<<<dim3((unsigned)g), dim3(256), 0, stream>>>(dstI, hist, E);
  }
  {
    int ntiles = N / 16;
    int grid = (ntiles + 3) / 4;
    node_qkv_kernel<<<dim3(grid), dim3(128), 0, stream>>>(
        nf, Wn, bn, Wq, bq, Wk, bk, Wv, bv, h, Qb, Kb, Vb, ntiles);
  }
  {
    int etiles = E / 16;
    int grid = (etiles + 3) / 4;
    edge_score_kernel<<<dim3(grid), dim3(128), 0, stream>>>(
        ef, srcI, dstI, We, be, Wea, bea, Qb, Kb, sc, mbuf, etiles);
  }
  {
    long g = (EH + 255) / 256;
    softmax_exp_kernel<<<dim3((unsigned)g), dim3(256), 0, stream>>>(
        srcI, sc, mbuf, den, EH);
    normalize_kernel<<<dim3((unsigned)g), dim3(256), 0, stream>>>(srcI, sc,
                                                                  den, EH);
  }
  scan_kernel<<<dim3(1), dim3(1024), 0, stream>>>(hist, offs, N);
  {
    long g = ((long)E + 255) / 256;
    build_csr_kernel<<<dim3((unsigned)g), dim3(256), 0, stream>>>(
        dstI, cursor, offs, eid, E);
  }
  {
    int grid = (N + 7) / 8;
    gather_out_kernel<<<dim3(grid), dim3(256), 0, stream>>>(eid, offs, srcI,
                                                            Vb, sc, out, N);
  }
  {
    int ntiles = N / 16;
    int grid = (ntiles + 3) / 4;
    out_proj_kernel<<<dim3(grid), dim3(128), 0, stream>>>(h, Wo, bo, out,
                                                          ntiles);
  }
}